// PureMambaEncoderBlock_58342835749240
// MI455X (gfx1250) — compile-verified
//
#include <hip/hip_runtime.h>
#include <hip/hip_bf16.h>
#include <math.h>

// ---------------------------------------------------------------------------
// MI455X / gfx1250 implementation of the PureMambaEncoderBlock reference.
// fp32 WMMA (v_wmma_f32_16x16x4_f32) for every GEMM; half-wave-per-channel
// sequential selective scan with cross-lane reduction; fused epilogues.
// Round 2: branch-free / compile-time-removed column masking in the GEMM
// main loop (previous version emitted s_and_saveexec diamonds per k-step).
// ---------------------------------------------------------------------------

typedef __attribute__((ext_vector_type(2))) float v2f;
typedef __attribute__((ext_vector_type(8))) float v8f;

static constexpr int Bn   = 2;     // batch
static constexpr int Dm   = 192;   // model dim
static constexpr int DINm = 384;   // inner dim
static constexpr int NST  = 16;    // state size
static constexpr int DTRm = 12;    // dt rank
static constexpr int L1   = 3136;  // 56*56
static constexpr int L2c  = 784;   // 28*28
static constexpr int PLD  = 48;    // proj buffer leading dim (44 padded to 48)

__device__ __forceinline__ float sigmoidf_(float x) { return 1.f / (1.f + __expf(-x)); }
__device__ __forceinline__ float siluf_(float x)    { return x * sigmoidf_(x); }
__device__ __forceinline__ float softplusf_(float x){ return (x > 20.f) ? x : log1pf(__expf(x)); }

// ------------------------------ WMMA GEMM ----------------------------------
// C[M,N] = A[M,K] @ W[N,K]^T (+ epilogue). One wave -> 16 rows x 32 cols.
enum { EP_NONE = 0, EP_BIAS = 1, EP_BIAS_RES = 2, EP_PROJBN = 3 };

template <int MODE, bool MASKED>
__global__ __launch_bounds__(128) void gemm_wmma_f32(
    const float* __restrict__ A, int lda,
    const float* __restrict__ W, int ldw,
    float* __restrict__ C, int ldc,
    int mtiles, int ngroups, int nwload, int nstore, int K,
    const float* __restrict__ bias, const float* __restrict__ res,
    const float* __restrict__ bng, const float* __restrict__ bnb,
    const float* __restrict__ bnm, const float* __restrict__ bnv)
{
  int wid = blockIdx.x * (blockDim.x >> 5) + (threadIdx.x >> 5);
  if (wid >= mtiles * ngroups) return;   // uniform per wave -> EXEC all-1 for WMMA
  int mt = wid % mtiles;
  int ng = wid / mtiles;
  int lane = threadIdx.x & 31;
  int half = lane >> 4;            // 0: K pair {k,k+1}; 1: {k+2,k+3}
  int l16  = lane & 15;
  int m0 = mt * 16, n0 = ng * 32;

  const float* Arow = A + (size_t)(m0 + l16) * lda + half * 2;
  int na = n0 + l16, nb = n0 + 16 + l16;
  // clamp W rows to a valid row; mask via select (no EXEC diamonds)
  int nac = MASKED ? ((na < nwload) ? na : (nwload - 1)) : na;
  int nbc = MASKED ? ((nb < nwload) ? nb : (nwload - 1)) : nb;
  const float* Wa = W + (size_t)nac * ldw + half * 2;
  const float* Wb = W + (size_t)nbc * ldw + half * 2;
  bool oka = na < nwload, okb = nb < nwload;
  v2f zf; zf.x = 0.f; zf.y = 0.f;

  v8f acc0 = {}; v8f acc1 = {};
  for (int k = 0; k < K; k += 4) {
    v2f af  = *(const v2f*)(Arow + k);
    v2f bf0 = *(const v2f*)(Wa + k);
    v2f bf1 = *(const v2f*)(Wb + k);
    if (MASKED) {           // lowers to v_cndmask, stays straight-line
      bf0 = oka ? bf0 : zf;
      bf1 = okb ? bf1 : zf;
    }
    acc0 = __builtin_amdgcn_wmma_f32_16x16x4_f32(false, af, false, bf0, (short)0, acc0, false, false);
    acc1 = __builtin_amdgcn_wmma_f32_16x16x4_f32(false, af, false, bf1, (short)0, acc1, false, false);
  }

  int rbase = m0 + half * 8;
  int c0 = n0 + l16, c1 = n0 + 16 + l16;
#pragma unroll
  for (int j = 0; j < 8; j++) {
    int r = rbase + j;
    float v0 = acc0[j], v1 = acc1[j];
    if (MODE == EP_BIAS || MODE == EP_BIAS_RES) {
      v0 += bias[c0];
      if (!MASKED || c1 < nstore) v1 += bias[c1];
    }
    if (MODE == EP_BIAS_RES) {
      v0 += res[(size_t)r * ldc + c0];
      v1 += res[(size_t)r * ldc + c1];
    }
    if (MODE == EP_PROJBN) {
      float s0 = bng[c0] * rsqrtf(bnv[c0] + 1e-5f);
      v0 = (v0 + bias[c0] - bnm[c0]) * s0 + bnb[c0];
      float s1 = bng[c1] * rsqrtf(bnv[c1] + 1e-5f);
      v1 = (v1 + bias[c1] - bnm[c1]) * s1 + bnb[c1];
    }
    if (!MASKED || c0 < nstore) C[(size_t)r * ldc + c0] = v0;
    if (!MASKED || c1 < nstore) C[(size_t)r * ldc + c1] = v1;
  }
}

// ------------------------------ LayerNorm ----------------------------------
// one wave per (b,l) row of 192 elements
__global__ __launch_bounds__(256) void ln_kernel(
    const float* __restrict__ x, const float* __restrict__ g,
    const float* __restrict__ b, float* __restrict__ h, int rows)
{
  int row = blockIdx.x * 8 + (threadIdx.x >> 5);
  if (row >= rows) return;
  int lane = threadIdx.x & 31;
  const float* rp = x + (size_t)row * Dm;
  float v[6]; float s = 0.f;
#pragma unroll
  for (int j = 0; j < 6; j++) { v[j] = rp[lane + 32 * j]; s += v[j]; }
#pragma unroll
  for (int off = 16; off > 0; off >>= 1) s += __shfl_xor(s, off, 32);
  float mu = s * (1.f / Dm);
  float q = 0.f;
#pragma unroll
  for (int j = 0; j < 6; j++) { float d = v[j] - mu; q += d * d; }
#pragma unroll
  for (int off = 16; off > 0; off >>= 1) q += __shfl_xor(q, off, 32);
  float rstd = rsqrtf(q * (1.f / Dm) + 1e-5f);
  float* hp = h + (size_t)row * Dm;
#pragma unroll
  for (int j = 0; j < 6; j++) {
    int c = lane + 32 * j;
    hp[c] = (v[j] - mu) * rstd * g[c] + b[c];
  }
}

// --------------------- depthwise causal conv (K=4) + SiLU ------------------
__global__ __launch_bounds__(256) void conv_silu_kernel(
    const float* __restrict__ xz, const float* __restrict__ cw,
    const float* __restrict__ cb, float* __restrict__ xi, int Lb, int n)
{
  int i = blockIdx.x * blockDim.x + threadIdx.x;
  if (i >= n) return;
  int e = i % DINm;
  int t = i / DINm;
  int l = t % Lb;
  int b = t / Lb;
  const float* base = xz + (size_t)b * Lb * 2 * DINm + e;
  float acc = cb[e];
#pragma unroll
  for (int j = 0; j < 4; j++) {
    int ll = l - 3 + j;
    if (ll >= 0) acc += cw[e * 4 + j] * base[(size_t)ll * 2 * DINm];
  }
  xi[(size_t)(b * Lb + l) * DINm + e] = siluf_(acc);
}

// ------------------------ dt = softplus(dtl @ Wdt^T + bdt) -----------------
__global__ __launch_bounds__(256) void dt_kernel(
    const float* __restrict__ proj, const float* __restrict__ Wdt,
    const float* __restrict__ bdt, float* __restrict__ dt, int Lb, int n)
{
  int i = blockIdx.x * blockDim.x + threadIdx.x;
  if (i >= n) return;
  int e = i % DINm;
  int t = i / DINm;
  const float* pr = proj + (size_t)t * PLD;
  float acc = bdt[e];
#pragma unroll
  for (int r = 0; r < DTRm; r++) acc += pr[r] * Wdt[e * DTRm + r];
  dt[(size_t)t * DINm + e] = softplusf_(acc);
}

// ---------------------------- selective scan -------------------------------
// half-wave (16 lanes = 16 states) per (b,e); sequential over L.
__global__ __launch_bounds__(256) void scan_kernel(
    const float* __restrict__ dt, const float* __restrict__ xi,
    const float* __restrict__ proj, const float* __restrict__ xz,
    const float* __restrict__ A_log, const float* __restrict__ Dskip,
    float* __restrict__ y, int Lb)
{
  int ch = blockIdx.x * 16 + (threadIdx.x >> 4);  // b*DIN + e, total 768
  int n = threadIdx.x & 15;
  if (ch >= Bn * DINm) return;
  int b = ch / DINm, e = ch % DINm;
  float Aen = -__expf(A_log[e * NST + n]);
  float dsk = Dskip[e];
  float hc = 0.f;
  size_t b384 = (size_t)b * Lb * DINm + e;
  size_t b768 = (size_t)b * Lb * 2 * DINm + DINm + e;   // z half
  size_t b48  = (size_t)b * Lb * PLD;
  for (int l = 0; l < Lb; l++) {
    float dtv = dt[b384 + (size_t)l * DINm];
    float xiv = xi[b384 + (size_t)l * DINm];
    float Bv = proj[b48 + (size_t)l * PLD + DTRm + n];
    float Cv = proj[b48 + (size_t)l * PLD + DTRm + NST + n];
    hc = __expf(dtv * Aen) * hc + (dtv * xiv) * Bv;
    float p = hc * Cv;
    p += __shfl_xor(p, 1, 16);
    p += __shfl_xor(p, 2, 16);
    p += __shfl_xor(p, 4, 16);
    p += __shfl_xor(p, 8, 16);
    if (n == 0) {
      float zv = xz[b768 + (size_t)l * 2 * DINm];
      y[b384 + (size_t)l * DINm] = (p + dsk * xiv) * siluf_(zv);
    }
  }
}

// -------------------------- layout helper kernels --------------------------
__global__ __launch_bounds__(256) void transpose_in_kernel(
    const float* __restrict__ x, float* __restrict__ xt, int n)
{ // (B,96,L1) -> (B*L1, 96)
  int i = blockIdx.x * blockDim.x + threadIdx.x;
  if (i >= n) return;
  int c = i % 96; int t = i / 96; int l = t % L1; int b = t / L1;
  xt[i] = x[((size_t)b * 96 + c) * L1 + l];
}

__global__ __launch_bounds__(256) void seq_to_chw_kernel(
    const float* __restrict__ seq, float* __restrict__ out, int Lb, int n)
{ // (B,Lb,D) -> (B,D,Lb)
  int i = blockIdx.x * blockDim.x + threadIdx.x;
  if (i >= n) return;
  int l = i % Lb; int t = i / Lb; int d = t % Dm; int b = t / Dm;
  out[i] = seq[((size_t)b * Lb + l) * Dm + d];
}

__global__ __launch_bounds__(256) void maxpool_kernel(
    const float* __restrict__ seq, float* __restrict__ pooled, int n)
{ // (B,3136,D) 56x56 -> (B,784,D) 28x28
  int i = blockIdx.x * blockDim.x + threadIdx.x;
  if (i >= n) return;
  int d = i % Dm; int t = i / Dm; int lp = t % L2c; int b = t / L2c;
  int hp = lp / 28, wp = lp % 28;
  int l0 = (hp * 2) * 56 + wp * 2;
  size_t base = (size_t)b * L1 * Dm + d;
  float a = seq[base + (size_t)l0 * Dm];
  float c = seq[base + (size_t)(l0 + 1) * Dm];
  float e = seq[base + (size_t)(l0 + 56) * Dm];
  float f = seq[base + (size_t)(l0 + 57) * Dm];
  pooled[i] = fmaxf(fmaxf(a, c), fmaxf(e, f));
}

// ------------------------------ host side ----------------------------------
static inline int cdiv_(long a, long b) { return (int)((a + b - 1) / b); }

static void launch_gemm(int mode, bool masked, const float* A, int lda,
                        const float* W, int ldw, float* C, int ldc, int M,
                        int ngroups, int nwload, int nstore, int K,
                        const float* bias, const float* res,
                        const float* bng, const float* bnb, const float* bnm,
                        const float* bnv, hipStream_t s)
{
  int mtiles = M / 16;
  int blocks = cdiv_((long)mtiles * ngroups, 4);
  if (masked) {
    gemm_wmma_f32<EP_NONE, true><<<blocks, 128, 0, s>>>(A, lda, W, ldw, C, ldc,
        mtiles, ngroups, nwload, nstore, K, bias, res, bng, bnb, bnm, bnv);
    return;
  }
  switch (mode) {
    case EP_NONE:
      gemm_wmma_f32<EP_NONE, false><<<blocks, 128, 0, s>>>(A, lda, W, ldw, C, ldc,
          mtiles, ngroups, nwload, nstore, K, bias, res, bng, bnb, bnm, bnv); break;
    case EP_BIAS:
      gemm_wmma_f32<EP_BIAS, false><<<blocks, 128, 0, s>>>(A, lda, W, ldw, C, ldc,
          mtiles, ngroups, nwload, nstore, K, bias, res, bng, bnb, bnm, bnv); break;
    case EP_BIAS_RES:
      gemm_wmma_f32<EP_BIAS_RES, false><<<blocks, 128, 0, s>>>(A, lda, W, ldw, C, ldc,
          mtiles, ngroups, nwload, nstore, K, bias, res, bng, bnb, bnm, bnv); break;
    default:
      gemm_wmma_f32<EP_PROJBN, false><<<blocks, 128, 0, s>>>(A, lda, W, ldw, C, ldc,
          mtiles, ngroups, nwload, nstore, K, bias, res, bng, bnb, bnm, bnv); break;
  }
}

extern "C" void kernel_launch(void* const* d_in, const int* in_sizes, int n_in,
                              void* d_out, int out_size, void* d_ws, size_t ws_size,
                              hipStream_t stream)
{
  (void)in_sizes; (void)n_in; (void)out_size; (void)ws_size;
  const float* x      = (const float*)d_in[0];
  const float* proj_w = (const float*)d_in[1];
  const float* proj_b = (const float*)d_in[2];
  const float* bn_g   = (const float*)d_in[3];
  const float* bn_b   = (const float*)d_in[4];
  const float* bn_mean= (const float*)d_in[5];
  const float* bn_var = (const float*)d_in[6];
  const float* ln_g   = (const float*)d_in[7];
  const float* ln_b   = (const float*)d_in[8];
  const float* Win    = (const float*)d_in[9];
  const float* b_in   = (const float*)d_in[10];
  const float* conv_w = (const float*)d_in[11];
  const float* conv_b = (const float*)d_in[12];
  const float* Wx     = (const float*)d_in[13];
  const float* Wdt    = (const float*)d_in[14];
  const float* bdt    = (const float*)d_in[15];
  const float* A_log  = (const float*)d_in[16];
  const float* Dskip  = (const float*)d_in[17];
  const float* Wout   = (const float*)d_in[18];
  const float* bout   = (const float*)d_in[19];
  float* out = (float*)d_out;
  float* ws  = (float*)d_ws;

  // workspace carving (floats); lifetime-based aliasing
  float* xseq  = ws;                       // B*L1*D      = 1204224 (persistent)
  float* hy    = xseq + 1204224;           // max(h,y)    = 2408448
  float* xzb   = hy + 2408448;             // B*L1*768    = 4816896
  float* xib   = xzb + 4816896;            // B*L1*384    = 2408448
  float* projb = xib + 2408448;            // B*L1*48     =  301056
  float* dtb   = projb + 301056;           // B*L1*384    = 2408448
  float* pool  = dtb + 2408448;            // B*L2*D      =  301056
  float* xt    = xzb;                      // alias: B*L1*96 used before xz
  float* oseq  = xzb;                      // alias: block-2 output, xz dead then

  const int TB = 256;

  // ---- stage 0: 1x1 conv proj + BN (fused epilogue) ----
  transpose_in_kernel<<<cdiv_(Bn * L1 * 96, TB), TB, 0, stream>>>(x, xt, Bn * L1 * 96);
  launch_gemm(EP_PROJBN, false, xt, 96, proj_w, 96, xseq, Dm, Bn * L1, Dm / 32, Dm,
              Dm, 96, proj_b, nullptr, bn_g, bn_b, bn_mean, bn_var, stream);

  // ---- mamba blocks 0,1 at L=3136 with residual ----
  for (int i = 0; i < 2; i++) {
    int rows = Bn * L1;
    ln_kernel<<<cdiv_(rows, 8), 256, 0, stream>>>(xseq, ln_g + i * Dm, ln_b + i * Dm, hy, rows);
    launch_gemm(EP_BIAS, false, hy, Dm, Win + (size_t)i * 2 * DINm * Dm, Dm, xzb,
                2 * DINm, rows, (2 * DINm) / 32, 2 * DINm, 2 * DINm, Dm,
                b_in + i * 2 * DINm, nullptr, nullptr, nullptr, nullptr, nullptr, stream);
    conv_silu_kernel<<<cdiv_((long)rows * DINm, TB), TB, 0, stream>>>(
        xzb, conv_w + i * DINm * 4, conv_b + i * DINm, xib, L1, rows * DINm);
    launch_gemm(EP_NONE, true, xib, DINm, Wx + (size_t)i * 44 * DINm, DINm, projb,
                PLD, rows, 2 /*64 cols padded*/, 44, PLD, DINm,
                nullptr, nullptr, nullptr, nullptr, nullptr, nullptr, stream);
    dt_kernel<<<cdiv_((long)rows * DINm, TB), TB, 0, stream>>>(
        projb, Wdt + i * DINm * DTRm, bdt + i * DINm, dtb, L1, rows * DINm);
    scan_kernel<<<(Bn * DINm) / 16, 256, 0, stream>>>(
        dtb, xib, projb, xzb, A_log + i * DINm * NST, Dskip + i * DINm, hy, L1);
    launch_gemm(EP_BIAS_RES, false, hy, DINm, Wout + (size_t)i * Dm * DINm, DINm,
                xseq, Dm, rows, Dm / 32, Dm, Dm, DINm,
                bout + i * Dm, xseq, nullptr, nullptr, nullptr, nullptr, stream);
  }

  // ---- skip output: (B,L1,D) -> (B,D,56,56) at out+301056 ----
  seq_to_chw_kernel<<<cdiv_(Bn * Dm * L1, TB), TB, 0, stream>>>(
      xseq, out + (size_t)Bn * Dm * L2c, L1, Bn * Dm * L1);

  // ---- 2x2 maxpool -> (B, 784, D) ----
  maxpool_kernel<<<cdiv_(Bn * L2c * Dm, TB), TB, 0, stream>>>(xseq, pool, Bn * L2c * Dm);

  // ---- mamba block 2 at L=784, no residual ----
  {
    int i = 2;
    int rows = Bn * L2c;
    ln_kernel<<<cdiv_(rows, 8), 256, 0, stream>>>(pool, ln_g + i * Dm, ln_b + i * Dm, hy, rows);
    launch_gemm(EP_BIAS, false, hy, Dm, Win + (size_t)i * 2 * DINm * Dm, Dm, xzb,
                2 * DINm, rows, (2 * DINm) / 32, 2 * DINm, 2 * DINm, Dm,
                b_in + i * 2 * DINm, nullptr, nullptr, nullptr, nullptr, nullptr, stream);
    conv_silu_kernel<<<cdiv_((long)rows * DINm, TB), TB, 0, stream>>>(
        xzb, conv_w + i * DINm * 4, conv_b + i * DINm, xib, L2c, rows * DINm);
    launch_gemm(EP_NONE, true, xib, DINm, Wx + (size_t)i * 44 * DINm, DINm, projb,
                PLD, rows, 2, 44, PLD, DINm,
                nullptr, nullptr, nullptr, nullptr, nullptr, nullptr, stream);
    dt_kernel<<<cdiv_((long)rows * DINm, TB), TB, 0, stream>>>(
        projb, Wdt + i * DINm * DTRm, bdt + i * DINm, dtb, L2c, rows * DINm);
    scan_kernel<<<(Bn * DINm) / 16, 256, 0, stream>>>(
        dtb, xib, projb, xzb, A_log + i * DINm * NST, Dskip + i * DINm, hy, L2c);
    launch_gemm(EP_BIAS, false, hy, DINm, Wout + (size_t)i * Dm * DINm, DINm, oseq,
                Dm, rows, Dm / 32, Dm, Dm, DINm,
                bout + i * Dm, nullptr, nullptr, nullptr, nullptr, nullptr, stream);
  }

  // ---- final output: (B,784,D) -> (B,D,28,28) at out+0 ----
  seq_to_chw_kernel<<<cdiv_(Bn * Dm * L2c, TB), TB, 0, stream>>>(
      oseq, out, L2c, Bn * Dm * L2c);
}